// MCPBRNN_Generic_PETconstraint_MIoutputloss_53695681134753
// MI455X (gfx1250) — compile-verified
//
#include <hip/hip_runtime.h>
#include <cstdint>

// ---------------------------------------------------------------------------
// MCPBRNN scan kernel for gfx1250 (MI455X).
// Sequential recurrence over n=32768 steps; one wave (32 lanes), wave32.
//   - lanes compute the G=8 hidden units of both gate networks in parallel
//   - 3-stage DPP butterfly (quad_perm x2 + row_half_mirror) reduces the
//     8-wide dot products entirely in the VALU (no DS latency)
//   - loop rotated: partial pre-activations (hxp1/u2 terms) computed in the
//     previous iteration, so the carry chain is c0 -> 1 fma -> selu
//   - input x[] streamed global->LDS with CDNA5 async-to-LDS instructions,
//     4-buffer ring, 2-chunk lookahead, gated by s_wait_asynccnt
//   - lanes 0..9 store the 10 output streams
// ---------------------------------------------------------------------------

#define CH_STEPS   512                 // steps per chunk
#define CH_BYTES   4096                // 512 steps * 2 floats * 4B
#define N_BUF      4                   // LDS ring depth

#if __has_builtin(__builtin_amdgcn_s_wait_asynccnt)
#define WAIT_ASYNC(N) __builtin_amdgcn_s_wait_asynccnt(N)
#else
#define WAIT_ASYNC(N) asm volatile("s_wait_asynccnt %0" :: "i"(N) : "memory")
#endif

#define ASYNC_LOAD_B128(lds32, gaddr64)                                        \
  asm volatile("global_load_async_to_lds_b128 %0, %1, off"                     \
               :: "v"(lds32), "v"(gaddr64) : "memory")

// ---- 8-lane butterfly sum ------------------------------------------------
#if __has_builtin(__builtin_amdgcn_update_dpp)
template <int CTRL>
__device__ __forceinline__ float dpp_add(float v) {
  int p = __builtin_amdgcn_update_dpp(0, __float_as_int(v), CTRL,
                                      0xf, 0xf, true);
  return v + __int_as_float(p);
}
__device__ __forceinline__ float reduce8(float v) {
  v = dpp_add<0xB1>(v);   // quad_perm [1,0,3,2]  : xor 1
  v = dpp_add<0x4E>(v);   // quad_perm [2,3,0,1]  : xor 2
  v = dpp_add<0x141>(v);  // row_half_mirror      : pairs opposite quads in 8
  return v;
}
#else
__device__ __forceinline__ float reduce8(float v) {
  v += __shfl_xor(v, 1, 32);
  v += __shfl_xor(v, 2, 32);
  v += __shfl_xor(v, 4, 32);
  return v;
}
#endif

__device__ __forceinline__ float selu_f(float x) {
  const float sc = 1.0507009873554805f;   // jax.nn.selu scale
  const float al = 1.6732632423543772f;   // jax.nn.selu alpha
  float neg = sc * (al * expm1f(x));      // matches jax (expm1-based)
  float pos = sc * x;
  return x > 0.0f ? pos : neg;
}

__device__ __forceinline__ float sigmoid_f(float x) {
  // numerically stable logistic; reciprocal via v_rcp_f32 (~1 ulp)
  float e = expf(-fabsf(x));
#if __has_builtin(__builtin_amdgcn_rcpf)
  float r = __builtin_amdgcn_rcpf(1.0f + e);
#else
  float r = 1.0f / (1.0f + e);
#endif
  return (x >= 0.0f) ? r : e * r;
}

__global__ __launch_bounds__(32)
void mcpbrnn_scan_kernel(const float* __restrict__ xg,        // (n,1,2)
                         const float* __restrict__ cmean,     // (1,)
                         const float* __restrict__ cstd,      // (1,)
                         const float* __restrict__ wr_yom,    // (1,1)
                         const float* __restrict__ wr_ylm,    // (1,1)
                         const float* __restrict__ wr_yfm,    // (1,1)
                         const float* __restrict__ bias_yom,  // (1,)
                         const float* __restrict__ wb1_yom,   // (2,8)
                         const float* __restrict__ wb2_yom,   // (8,1)
                         const float* __restrict__ rb_o,      // (1,8)
                         const float* __restrict__ b_ln_yom,  // (1,1)
                         const float* __restrict__ bias_ylm,  // (1,)
                         const float* __restrict__ wb1_ylm,   // (2,8)
                         const float* __restrict__ wb2_ylm,   // (8,1)
                         const float* __restrict__ rb_l,      // (1,8)
                         const float* __restrict__ b_ln_ylm,  // (1,1)
                         const int*   __restrict__ time_lag_p,
                         float* __restrict__ out,             // 10 * n
                         int n)
{
  __shared__ __align__(16) float xs[N_BUF * CH_BYTES / 4];   // 16 KB LDS ring

  const int lane = threadIdx.x;       // 0..31 (wave32)
  const int h    = lane & 7;          // hidden-unit index (replicated x4)

  // ---- uniform parameters -------------------------------------------------
  const float mo     = cmean[0];
  const float so     = cstd[0];
  const float inv_so = 1.0f / so;     // cstd==1 in harness -> bit-exact
  const float mlc    = 2.9086f;       // ML
  const float inv_sl = 1.0f / 1.898f; // 1/SL
  const float e_o = expf(wr_yom[0]);
  const float e_l = expf(wr_ylm[0]);
  const float e_f = expf(wr_yfm[0]);
  const float den_r = e_o + e_l + e_f;
  const float oo1 = e_o / den_r;
  const float ol1 = e_l / den_r;
  const float bln_o = b_ln_yom[0];
  const float bln_l = b_ln_ylm[0];
  const int   tlag  = time_lag_p[0];

  // ---- per-lane weights (hidden unit h of both networks) ------------------
  const float w1o0 = wb1_yom[h];
  const float w1o1 = wb1_yom[8 + h];
  const float w2o  = wb2_yom[h];
  const float w1l0 = wb1_ylm[h];
  const float w1l1 = wb1_ylm[8 + h];
  const float w2l  = wb2_ylm[h];
  const float cb_o = bias_yom[0] - rb_o[h];   // folded bias - relu_bias
  const float cb_l = bias_ylm[0] - rb_l[h];

  float* outp = out + (size_t)lane * (size_t)n;   // lanes 0..9 own a stream

  // ---- async chunk loader -------------------------------------------------
  const uint32_t xs_base = (uint32_t)(uintptr_t)(&xs[0]);
  auto issue_chunk = [&](int c) {
    const char* g = (const char*)xg + ((size_t)c << 12) + ((size_t)lane << 4);
    uint32_t    l = xs_base + ((uint32_t)(c & (N_BUF - 1)) << 12) +
                    ((uint32_t)lane << 4);
#pragma unroll
    for (int i = 0; i < 8; ++i) {     // 8 rows * 32 lanes * 16B = 4096B
      uint64_t ga = (uint64_t)(uintptr_t)(g + (i << 9));
      ASYNC_LOAD_B128(l + (uint32_t)(i << 9), ga);
    }
  };

  // ---- recurrence state ---------------------------------------------------
  float c0   = 0.0f;
  float hxp1 = 0.0f;
  // part_o for step b computed at end of step b-1 (b=0: c0p1 = 0 = hxp1 init)
  float part_o = cb_o + ((0.0f - mo) * inv_so) * w1o1;

  // step: uses carried part_o and caller-supplied part_l for step b;
  // updates c0/hxp1 and computes part_o for step b+1 (off critical path).
  auto stepc = [&](int b, float u1, float u2, float pl) {
    const float a0 = (c0 - mo) * inv_so;          // carry-critical entry

    const float pre_o = part_o + a0 * w1o0;       // 1 fma from c0 to selu
    const float pre_l = pl     + a0 * w1l0;
    float so_ = reduce8(selu_f(pre_o) * w2o);
    float sl_ = reduce8(selu_f(pre_l) * w2l);

    const float oo = oo1 * sigmoid_f(bln_o + so_);
    const float ol = ol1 * sigmoid_f(bln_l + sl_);

    const bool  pos   = c0 > 0.0f;
    const float cden  = pos ? c0 : 1.0f;
    const float ratio = u2 / cden;                // hidden under sigmoid chain
    const float olc   = pos ? (ol - fmaxf(ol - ratio, 0.0f)) : ol;
    const float f     = 1.0f - oo - olc;

    // per-lane output select: (oo*c0, c0, ol*c0, olc*c0, 0, 0, oo, ol, olc, f)
    float v = c0;                       // lane 1
    v = (lane == 0) ? oo  * c0 : v;
    v = (lane == 2) ? ol  * c0 : v;
    v = (lane == 3) ? olc * c0 : v;
    v = (lane == 4) ? 0.0f     : v;
    v = (lane == 5) ? 0.0f     : v;
    v = (lane == 6) ? oo       : v;
    v = (lane == 7) ? ol       : v;
    v = (lane == 8) ? olc      : v;
    v = (lane == 9) ? f        : v;
    if (lane < 10) outp[b] = v;

    // carry update
    const bool use_lag = (b >= tlag);
    hxp1 = use_lag ? c0 : hxp1;
    c0   = f * c0 + u1;

    // part_o for step b+1 (depends only on updated hxp1)
    const bool  use_lag_n = (b + 1 >= tlag);
    const float c0p1n     = use_lag_n ? hxp1 : 0.0f;
    part_o = cb_o + ((c0p1n - mo) * inv_so) * w1o1;
  };

  // ---- pipelined main loop over full chunks -------------------------------
  const int nfull = n / CH_STEPS;
  int b = 0;
  if (nfull > 0) {
    issue_chunk(0);
    if (nfull > 1) issue_chunk(1);
    for (int c = 0; c < nfull; ++c) {
      if (c + 2 < nfull) {
        issue_chunk(c + 2);
        WAIT_ASYNC(16);               // <=16 outstanding -> chunk c complete
      } else if (c + 1 < nfull) {
        WAIT_ASYNC(8);                // only chunk c+1 may remain in flight
      } else {
        WAIT_ASYNC(0);
      }
      const float2* xb =
          (const float2*)(xs + ((size_t)(c & (N_BUF - 1)) << 10));
      // software pipeline: next-step LDS read + part_l one step ahead
      float2 uv = xb[0];
      float  pl = cb_l + ((uv.y - mlc) * inv_sl) * w1l1;  // chunk-head fixup
      for (int s = 0; s < CH_STEPS; ++s, ++b) {
        const int    sn  = (s + 1 < CH_STEPS) ? (s + 1) : s;
        const float2 un  = xb[sn];
        const float  pln = cb_l + ((un.y - mlc) * inv_sl) * w1l1;
        stepc(b, uv.x, uv.y, pl);
        uv = un;
        pl = pln;   // discarded at chunk boundary (recomputed at next head)
      }
    }
  }
  // remainder steps (n % 512) straight from global (none for n = 32768)
  for (; b < n; ++b) {
    const float u1 = xg[2 * b];
    const float u2 = xg[2 * b + 1];
    const float pl = cb_l + ((u2 - mlc) * inv_sl) * w1l1;
    stepc(b, u1, u2, pl);
  }
}

extern "C" void kernel_launch(void* const* d_in, const int* in_sizes, int n_in,
                              void* d_out, int out_size, void* d_ws, size_t ws_size,
                              hipStream_t stream) {
  (void)n_in; (void)d_ws; (void)ws_size; (void)out_size;

  const float* x        = (const float*)d_in[0];
  // d_in[1] = y_obs (unused by the forward pass)
  const float* cmean    = (const float*)d_in[2];
  const float* cstd     = (const float*)d_in[3];
  const float* wr_yom   = (const float*)d_in[4];
  const float* wr_ylm   = (const float*)d_in[5];
  const float* wr_yfm   = (const float*)d_in[6];
  const float* bias_yom = (const float*)d_in[7];
  const float* wb1_yom  = (const float*)d_in[8];
  const float* wb2_yom  = (const float*)d_in[9];
  const float* rb_o     = (const float*)d_in[10];
  const float* b_ln_yom = (const float*)d_in[11];
  const float* bias_ylm = (const float*)d_in[12];
  const float* wb1_ylm  = (const float*)d_in[13];
  const float* wb2_ylm  = (const float*)d_in[14];
  const float* rb_l     = (const float*)d_in[15];
  const float* b_ln_ylm = (const float*)d_in[16];
  // d_in[17] = epoch (unused)
  const int*   time_lag = (const int*)d_in[18];

  const int n = in_sizes[0] / 2;      // x is (n,1,2)

  mcpbrnn_scan_kernel<<<dim3(1), dim3(32), 0, stream>>>(
      x, cmean, cstd, wr_yom, wr_ylm, wr_yfm, bias_yom, wb1_yom, wb2_yom,
      rb_o, b_ln_yom, bias_ylm, wb1_ylm, wb2_ylm, rb_l, b_ln_ylm,
      time_lag, (float*)d_out, n);
}